// PMF_46059229282319
// MI455X (gfx1250) — compile-verified
//
#include <hip/hip_runtime.h>
#include <cstdint>
#include <cstddef>

// PMF gather-multiply-sum, MI455X (gfx1250) tuned.
//
// Memory-bound (0.19 FLOP/byte): the win is (a) keeping the 30MB U/V tables
// resident in the 192MB L2 while streaming idx/outputs past it with
// non-temporal hints, and (b) b128 vector memory ops everywhere.

typedef float v4f __attribute__((ext_vector_type(4)));
typedef int   v4i __attribute__((ext_vector_type(4)));

#define LATENT 5
#define IPT 4          // pairs per thread
#define BLOCK 256      // 8 wave32s per block

// 16-byte load with only 4-byte alignment guarantee (row stride = 20 bytes).
// Reading floats [0..3] of a 5-float row is always in-bounds.
__device__ __forceinline__ v4f load_row4(const float* __restrict__ p) {
    v4f r;
    __builtin_memcpy(&r, p, sizeof(r));   // align-4 16B load -> global_load_b128
    return r;
}

__global__ __launch_bounds__(BLOCK)
void pmf_gather_kernel(const float* __restrict__ U,
                       const float* __restrict__ V,
                       const int*   __restrict__ uidx,
                       const int*   __restrict__ vidx,
                       float*       __restrict__ uv,     // n x 3
                       float*       __restrict__ preds,  // n
                       int n)
{
    const int t    = blockIdx.x * BLOCK + threadIdx.x;
    const int base = t * IPT;

    if (base + IPT <= n) {
        // ---- streaming index loads: b128, non-temporal (don't pollute L2) ----
        const v4i ui = __builtin_nontemporal_load((const v4i*)(uidx + base));
        const v4i vi = __builtin_nontemporal_load((const v4i*)(vidx + base));

        // ---- gathers: default (RT) temporal hint -> tables stay L2-resident ----
        const v4f a0 = load_row4(U + (size_t)(uint32_t)ui.x * LATENT);
        const v4f a1 = load_row4(U + (size_t)(uint32_t)ui.y * LATENT);
        const v4f a2 = load_row4(U + (size_t)(uint32_t)ui.z * LATENT);
        const v4f a3 = load_row4(U + (size_t)(uint32_t)ui.w * LATENT);
        const v4f b0 = load_row4(V + (size_t)(uint32_t)vi.x * LATENT);
        const v4f b1 = load_row4(V + (size_t)(uint32_t)vi.y * LATENT);
        const v4f b2 = load_row4(V + (size_t)(uint32_t)vi.z * LATENT);
        const v4f b3 = load_row4(V + (size_t)(uint32_t)vi.w * LATENT);

        const float q0x = a0.x * b0.x, q0y = a0.y * b0.y, q0z = a0.z * b0.z;
        const float q1x = a1.x * b1.x, q1y = a1.y * b1.y, q1z = a1.z * b1.z;
        const float q2x = a2.x * b2.x, q2y = a2.y * b2.y, q2z = a2.z * b2.z;
        const float q3x = a3.x * b3.x, q3y = a3.y * b3.y, q3z = a3.z * b3.z;

        // ---- uv: 12 consecutive floats, 16B-aligned (base*12 bytes, base%4==0)
        //      3x global_store_b128 th:TH_NT, contiguous across the wave ----
        float* o = uv + (size_t)base * 3;
        const v4f w0 = { q0x, q0y, q0z, q1x };
        const v4f w1 = { q1y, q1z, q2x, q2y };
        const v4f w2 = { q2z, q3x, q3y, q3z };
        __builtin_nontemporal_store(w0, (v4f*)(o + 0));
        __builtin_nontemporal_store(w1, (v4f*)(o + 4));
        __builtin_nontemporal_store(w2, (v4f*)(o + 8));

        // ---- preds: one b128 NT store ----
        const v4f ps = { q0x + q0y + q0z,
                         q1x + q1y + q1z,
                         q2x + q2y + q2z,
                         q3x + q3y + q3z };
        __builtin_nontemporal_store(ps, (v4f*)(preds + base));
    } else {
        // tail (n not a multiple of IPT) — scalar, still coalesced enough
        for (int i = base; i < n; ++i) {
            const int   uu = uidx[i];
            const int   vv = vidx[i];
            const float* up = U + (size_t)(uint32_t)uu * LATENT;
            const float* vp = V + (size_t)(uint32_t)vv * LATENT;
            const float x = up[0] * vp[0];
            const float y = up[1] * vp[1];
            const float z = up[2] * vp[2];
            float* o = uv + (size_t)i * 3;
            __builtin_nontemporal_store(x, o + 0);
            __builtin_nontemporal_store(y, o + 1);
            __builtin_nontemporal_store(z, o + 2);
            __builtin_nontemporal_store(x + y + z, preds + i);
        }
    }
}

extern "C" void kernel_launch(void* const* d_in, const int* in_sizes, int n_in,
                              void* d_out, int out_size, void* d_ws, size_t ws_size,
                              hipStream_t stream)
{
    // setup_inputs() order: U, V, u_idx, v_idx
    const float* U    = (const float*)d_in[0];
    const float* V    = (const float*)d_in[1];
    const int*   uidx = (const int*)d_in[2];
    const int*   vidx = (const int*)d_in[3];
    const int    n    = in_sizes[2];          // N_PAIRS

    // d_out = [uv (n*3 f32) | preds (n f32)], reference return order
    float* uv    = (float*)d_out;
    float* preds = (float*)d_out + (size_t)n * 3;

    const int blocks = (n + BLOCK * IPT - 1) / (BLOCK * IPT);
    pmf_gather_kernel<<<blocks, BLOCK, 0, stream>>>(U, V, uidx, vidx, uv, preds, n);
}